// VectorizedSSM_69758858821939
// MI455X (gfx1250) — compile-verified
//
#include <hip/hip_runtime.h>
#include <hip/hip_bf16.h>
#include <math.h>

typedef __attribute__((ext_vector_type(2))) float v2f;
typedef __attribute__((ext_vector_type(8))) float v8f;

#define BB 2
#define LL 2048
#define DD 1024
#define NN 16
#define BL (BB * LL)

#define NBPB 4            // scan blocks per batch
#define SCAN_T 256        // threads per scan block (NBPB*SCAN_T == DD)
#define NSLOT ((LL + 9) / 10)   // 205 flagged steps
#define ACC_PAD 2048
#define CNT_PAD 1024

// ---- hardware transcendental helpers -------------------------------------
// v_tanh_f32 is a gfx1250 TRANS32 op; trailing v_nop satisfies the
// "1 independent op after TRANS before use" hazard (compiler can't see
// through the asm to schedule it).
__device__ __forceinline__ float tanh_hw(float x) {
    float r;
    asm volatile("v_tanh_f32 %0, %1\n\tv_nop" : "=v"(r) : "v"(x));
    return r;
}

__device__ __forceinline__ float softplus_f(float z) {
    // log(1+e^z); __expf/__logf lower to v_exp_f32/v_log_f32
    return (z > 20.f) ? z : __logf(1.f + __expf(z));
}

// soft_clamp with all scales folded into compile-time constants
// (avoids the precise-fdiv v_div_scale chain: sh/(r+1e-8) and r/sh are
// literal quotients, constant-folded by the compiler).
__device__ __forceinline__ float softclamp_f(float v, float mn, float mx, float sh) {
    const float c      = 0.5f * (mx + mn);
    const float r      = 0.5f * (mx - mn);
    const float win    = sh / (r + 1e-8f);   // constant-folded
    const float wout   = r / sh;             // constant-folded
    return tanh_hw((v - c) * win) * wout + c;
}

// ---------------------------------------------------------------------------
// dt = soft_clamp(softplus(x @ W_dt^T + b_dt)) via fp32 WMMA 16x16x4.
// ---------------------------------------------------------------------------
__global__ void __launch_bounds__(256) dt_proj_wmma(
    const float* __restrict__ x, const float* __restrict__ Wdt,
    const float* __restrict__ bdt, float* __restrict__ dt)
{
    const int wid  = (blockIdx.x * 256 + (int)threadIdx.x) >> 5;
    const int lane = threadIdx.x & 31;
    const int tiles_n = DD / 16;
    const int tm = wid / tiles_n;
    const int tn = wid % tiles_n;
    const int m  = lane & 15;
    const int kh = lane >> 4;

    const float* ap = x   + (size_t)(tm * 16 + m) * DD + 2 * kh;
    const float* bp = Wdt + (size_t)(tn * 16 + m) * DD + 2 * kh;

    v8f c = {};
#pragma unroll 8
    for (int k0 = 0; k0 < DD; k0 += 4) {
        v2f a = *(const v2f*)(ap + k0);
        v2f b = *(const v2f*)(bp + k0);
        c = __builtin_amdgcn_wmma_f32_16x16x4_f32(false, a, false, b,
                                                  (short)0, c, false, false);
    }

    const int col    = tn * 16 + m;
    const float bias = bdt[col];
#pragma unroll
    for (int i = 0; i < 8; ++i) {
        const int row = tm * 16 + i + 8 * kh;
        const float sp = softplus_f(c[i] + bias);
        dt[(size_t)row * DD + col] = softclamp_f(sp, 0.001f, 0.1f, 0.5f);
    }
}

// ---------------------------------------------------------------------------
// Bp = x @ W_B^T, Cp = x @ W_C^T; two accumulators share A-tile loads.
// ---------------------------------------------------------------------------
__global__ void __launch_bounds__(256) bc_proj_wmma(
    const float* __restrict__ x, const float* __restrict__ WB,
    const float* __restrict__ WC, float* __restrict__ Bp, float* __restrict__ Cp)
{
    const int wid  = (blockIdx.x * 256 + (int)threadIdx.x) >> 5;
    const int lane = threadIdx.x & 31;
    const int m  = lane & 15;
    const int kh = lane >> 4;

    const float* ap  = x  + (size_t)(wid * 16 + m) * DD + 2 * kh;
    const float* bbp = WB + (size_t)m * DD + 2 * kh;
    const float* bcp = WC + (size_t)m * DD + 2 * kh;

    v8f cb = {}, cc = {};
#pragma unroll 8
    for (int k0 = 0; k0 < DD; k0 += 4) {
        v2f a  = *(const v2f*)(ap  + k0);
        v2f b1 = *(const v2f*)(bbp + k0);
        v2f b2 = *(const v2f*)(bcp + k0);
        cb = __builtin_amdgcn_wmma_f32_16x16x4_f32(false, a, false, b1, (short)0, cb, false, false);
        cc = __builtin_amdgcn_wmma_f32_16x16x4_f32(false, a, false, b2, (short)0, cc, false, false);
    }

#pragma unroll
    for (int i = 0; i < 8; ++i) {
        const int row = wid * 16 + i + 8 * kh;
        Bp[row * NN + m] = cb[i];
        Cp[row * NN + m] = cc[i];
    }
}

// ---------------------------------------------------------------------------
// Zero the cross-block sync slots (every launch; graph-replay safe).
// ---------------------------------------------------------------------------
__global__ void __launch_bounds__(1024) init_sync(float* acc, unsigned* cnt)
{
    const int i = blockIdx.x * 1024 + threadIdx.x;
    if (i < ACC_PAD) acc[i] = 0.f;
    if (i < CNT_PAD) cnt[i] = 0u;
}

// ---------------------------------------------------------------------------
// Sequential scan, 4 blocks x 256 threads per batch. Thread owns h[d,0..15]
// in VGPRs. Periodic norm clamp = deterministic slot-barrier across the 4
// blocks of a batch (partials summed in fixed order).
// ---------------------------------------------------------------------------
__global__ void __launch_bounds__(SCAN_T) ssm_scan(
    const float* __restrict__ x,  const float* __restrict__ dt,
    const float* __restrict__ Bp, const float* __restrict__ Cp,
    const float* __restrict__ A_log, const float* __restrict__ Dparam,
    float* __restrict__ acc, unsigned* __restrict__ cnt,
    float* __restrict__ y)
{
    const int b    = blockIdx.x / NBPB;
    const int sub  = blockIdx.x % NBPB;
    const int d    = sub * SCAN_T + (int)threadIdx.x;
    const int lane = threadIdx.x & 31;
    const int wave = threadIdx.x >> 5;
    __shared__ float red[SCAN_T / 32];
    __shared__ float bcast;

    float A[NN], h[NN];
#pragma unroll
    for (int n = 0; n < NN; ++n) {
        // A = -exp(soft_clamp(A_log,-5,5,0.5)) = -exp(10*tanh(A_log/10))
        const float a = A_log[d * NN + n];
        A[n] = -__expf(10.f * tanh_hw(0.1f * a));
        h[n] = 0.f;
    }
    const float Dpar = Dparam[d];
    const size_t bbase = (size_t)b * LL;

    // double-buffered per-step operands
    float xv, dtv;
    float4 bv[4], cv[4];
    {
        const size_t r0 = bbase;
        xv  = x[r0 * DD + d];
        dtv = dt[r0 * DD + d];
        const float4* b4 = (const float4*)(Bp + r0 * NN);
        const float4* c4 = (const float4*)(Cp + r0 * NN);
#pragma unroll
        for (int i = 0; i < 4; ++i) { bv[i] = b4[i]; cv[i] = c4[i]; }
    }

    for (int t = 0; t < LL; ++t) {
        const size_t r = bbase + (size_t)t;

        // prefetch step t+1 while the trans chain of step t executes
        float xn = 0.f, dtn = 0.f;
        float4 bn[4], cn[4];
        if (t + 1 < LL) {
            const size_t rn = r + 1;
            xn  = x[rn * DD + d];
            dtn = dt[rn * DD + d];
            const float4* b4 = (const float4*)(Bp + rn * NN);
            const float4* c4 = (const float4*)(Cp + rn * NN);
#pragma unroll
            for (int i = 0; i < 4; ++i) { bn[i] = b4[i]; cn[i] = c4[i]; }
        }

        float Bl[NN], Cl[NN];
#pragma unroll
        for (int i = 0; i < 4; ++i) {
            Bl[4*i+0] = bv[i].x; Bl[4*i+1] = bv[i].y; Bl[4*i+2] = bv[i].z; Bl[4*i+3] = bv[i].w;
            Cl[4*i+0] = cv[i].x; Cl[4*i+1] = cv[i].y; Cl[4*i+2] = cv[i].z; Cl[4*i+3] = cv[i].w;
        }

        const float dbx = dtv * xv;
#pragma unroll
        for (int n = 0; n < NN; ++n) {
            const float dta = dtv * A[n];
            // soft_clamp(dta,-8,8,0.5) = 16*tanh(dta/16); dA = exp(that)
            const float th = tanh_hw(dta * 0.0625f);
            const float da = __expf(16.f * th);
            h[n] = fmaf(h[n], da, dbx * Bl[n]);
        }

        if ((t % 10) == 0) {
            float s = 0.f;
#pragma unroll
            for (int n = 0; n < NN; ++n) s = fmaf(h[n], h[n], s);
#pragma unroll
            for (int off = 16; off > 0; off >>= 1) s += __shfl_xor(s, off, 32);
            if (lane == 0) red[wave] = s;
            __syncthreads();
            if (threadIdx.x == 0) {
                float p = 0.f;
#pragma unroll
                for (int w = 0; w < SCAN_T / 32; ++w) p += red[w];
                const int slot = b * NSLOT + t / 10;
                __hip_atomic_store(&acc[slot * NBPB + sub], p,
                                   __ATOMIC_RELAXED, __HIP_MEMORY_SCOPE_AGENT);
                __hip_atomic_fetch_add(&cnt[slot], 1u,
                                       __ATOMIC_ACQ_REL, __HIP_MEMORY_SCOPE_AGENT);
                while (__hip_atomic_load(&cnt[slot], __ATOMIC_ACQUIRE,
                                         __HIP_MEMORY_SCOPE_AGENT) < NBPB) { }
                float tot = 0.f;
#pragma unroll
                for (int k = 0; k < NBPB; ++k)   // fixed order -> deterministic
                    tot += __hip_atomic_load(&acc[slot * NBPB + k],
                                             __ATOMIC_RELAXED, __HIP_MEMORY_SCOPE_AGENT);
                bcast = tot;
            }
            __syncthreads();
            const float norm  = sqrtf(bcast);
            const float scale = fminf(20.f / (norm + 1e-8f), 1.f);
#pragma unroll
            for (int n = 0; n < NN; ++n) h[n] *= scale;
            __syncthreads();   // protect red[]/bcast before next flagged step
        }

        float yv = Dpar * xv;
#pragma unroll
        for (int n = 0; n < NN; ++n) yv = fmaf(h[n], Cl[n], yv);
        y[r * DD + d] = yv;

        // rotate prefetched operands
        xv = xn; dtv = dtn;
#pragma unroll
        for (int i = 0; i < 4; ++i) { bv[i] = bn[i]; cv[i] = cn[i]; }
    }
}

extern "C" void kernel_launch(void* const* d_in, const int* in_sizes, int n_in,
                              void* d_out, int out_size, void* d_ws, size_t ws_size,
                              hipStream_t stream) {
    const float* x      = (const float*)d_in[0];
    const float* Wdt    = (const float*)d_in[1];
    const float* bdt    = (const float*)d_in[2];
    const float* WB     = (const float*)d_in[3];
    const float* WC     = (const float*)d_in[4];
    const float* A_log  = (const float*)d_in[5];
    const float* Dparam = (const float*)d_in[6];
    float* out = (float*)d_out;

    char* ws = (char*)d_ws;
    float* dt  = (float*)ws;                                       // BL*DD
    float* Bp  = (float*)(ws + (size_t)BL * DD * sizeof(float));   // BL*NN
    float* Cp  = Bp + (size_t)BL * NN;                             // BL*NN
    float* acc = Cp + (size_t)BL * NN;                             // ACC_PAD
    unsigned* cnt = (unsigned*)(acc + ACC_PAD);                    // CNT_PAD

    init_sync<<<2, 1024, 0, stream>>>(acc, cnt);
    dt_proj_wmma<<<2048, 256, 0, stream>>>(x, Wdt, bdt, dt);
    bc_proj_wmma<<<32, 256, 0, stream>>>(x, WB, WC, Bp, Cp);
    ssm_scan<<<BB * NBPB, SCAN_T, 0, stream>>>(x, dt, Bp, Cp, A_log, Dparam,
                                               acc, cnt, out);
}